// EMDLoss_9938554323206
// MI455X (gfx1250) — compile-verified
//
#include <hip/hip_runtime.h>

// EMD approx-match (Fan et al.) for B=8, N=M=2048, D=3 on gfx1250 (MI455X).
// d2 tiles generated on the fly with V_WMMA_F32_16X16X4_F32:
//   A row  = (-2x, -2y, -2z, 1),  B col = (x2x, x2y, x2z, |x2|^2)
//   => A*B = -2<x1,x2> + |x2|^2 ;  d2 = A*B + |x1|^2
// Per level: P1 rowsum -> R1 (rsinv) -> P2 colsum -> R2 (r, satr') ->
//            P3 (cost, rowsum2) -> R3 (satl').
// Blocks of 8 waves share one (batch, chunk); chunk operands staged in LDS.
// exp(level*d2) = raw v_exp_f32 of fma(lv2, wmma, lv2*n1), lv2 = level*log2(e);
// subnormal-range inputs flush to 0, which is the desired weight anyway.
// All reductions deterministic (per-wave partials, fixed-order combines).

typedef __attribute__((ext_vector_type(2))) float v2f;
typedef __attribute__((ext_vector_type(8))) float v8f;

#define Bsz 8
#define Nn  2048
#define Mm  2048
#define TN  (Nn / 16)       // 128 row strips per batch
#define TM  (Mm / 16)       // 128 col strips per batch
#define JC  4               // column chunks (pass 1/3)
#define IC  4               // row chunks (pass 2)
#define CH  (Mm / JC)       // 512 elements per chunk
#define BN  (Bsz * Nn)
#define BM  (Bsz * Mm)
#define WPB 8               // waves per 256-thread block
#define EPSV 1e-9f

__device__ __forceinline__ float redux16(float v) {
    v += __shfl_xor(v, 1);
    v += __shfl_xor(v, 2);
    v += __shfl_xor(v, 4);
    v += __shfl_xor(v, 8);
    return v;
}

__device__ __forceinline__ float fast_exp2(float x) {
    return __builtin_amdgcn_exp2f(x);      // raw v_exp_f32, no range fixup
}
__device__ __forceinline__ float fast_sqrt(float x) {
    return __builtin_amdgcn_sqrtf(x);      // raw v_sqrt_f32
}

__device__ __forceinline__ v8f wmma_d2(v2f a, v2f b) {
    v8f c = {};
    // D = A(16x4, f32) x B(4x16, f32) + 0
    c = __builtin_amdgcn_wmma_f32_16x16x4_f32(false, a, false, b,
                                              (short)0, c, false, false);
    return c;
}

// ---------------------------------------------------------------- prep
__global__ __launch_bounds__(256) void k_prep(const float* __restrict__ x1,
                                              const float* __restrict__ x2,
                                              float* __restrict__ n1,
                                              float* __restrict__ n2,
                                              float* __restrict__ satl,
                                              float* __restrict__ satr,
                                              float* __restrict__ costp) {
    int idx = blockIdx.x * 256 + threadIdx.x;     // 0..BN-1 (BN == BM)
    if (idx < BN) {
        const float* p = x1 + (size_t)idx * 3;
        n1[idx] = p[0] * p[0] + p[1] * p[1] + p[2] * p[2];
        satl[idx] = 1.0f;                         // factorl = max(N,M)/N = 1
    }
    if (idx < BM) {
        const float* p = x2 + (size_t)idx * 3;
        n2[idx] = p[0] * p[0] + p[1] * p[1] + p[2] * p[2];
        satr[idx] = 1.0f;                         // factorr = 1
    }
    if (idx < Bsz * TN * JC) costp[idx] = 0.0f;
}

// ---------------------------------------------------------------- pass 1: rowsum partials
// rowsum0[i] = sum_j exp(level*d2) * satr[j]
__global__ __launch_bounds__(256) void k_pass1(const float* __restrict__ x1,
                                               const float* __restrict__ x2,
                                               const float* __restrict__ n1,
                                               const float* __restrict__ n2,
                                               const float* __restrict__ satr,
                                               float* __restrict__ rowsum_part,
                                               float lv2) {
    __shared__ float sx[CH], sy[CH], sz[CH], sn[CH], ssr[CH];
    const int tid = threadIdx.x;
    const int g = blockIdx.x >> 4;                // (b, chunk) group
    const int sb = blockIdx.x & 15;
    const int b = g >> 2, chunk = g & 3;
    const size_t bN = (size_t)b * Nn;
    const size_t bM = (size_t)b * Mm;
    const int j0 = chunk * CH;

    for (int i = tid; i < CH; i += 256) {         // stage chunk operands
        const float* p = x2 + (bM + j0 + i) * 3;
        sx[i] = p[0]; sy[i] = p[1]; sz[i] = p[2];
        sn[i] = n2[bM + j0 + i];
        ssr[i] = satr[bM + j0 + i];
    }
    __syncthreads();

    const int lane = tid & 31;
    const int s = sb * WPB + (tid >> 5);          // strip 0..127
    const int row0 = s * 16;
    const int m16 = lane & 15, hf = lane >> 4;

    const float* p1 = x1 + (bN + row0 + m16) * 3;
    v2f a;
    a.x = hf ? -2.0f * p1[2] : -2.0f * p1[0];     // K0 / K2
    a.y = hf ? 1.0f          : -2.0f * p1[1];     // K1 / K3

    float ln1v[8];
#pragma unroll
    for (int v = 0; v < 8; ++v) ln1v[v] = lv2 * n1[bN + row0 + v + 8 * hf];

    float acc[8];
#pragma unroll
    for (int v = 0; v < 8; ++v) acc[v] = 0.0f;

    for (int t = 0; t < CH / 16; ++t) {
        int cc = t * 16 + m16;
        v2f bb;
        bb.x = hf ? sz[cc] : sx[cc];
        bb.y = hf ? sn[cc] : sy[cc];
        float sr = ssr[cc];
        v8f c = wmma_d2(a, bb);
#pragma unroll
        for (int v = 0; v < 8; ++v) {
            float e = fast_exp2(fmaf(lv2, c[v], ln1v[v]));   // exp(level*d2)
            acc[v] = fmaf(sr, e, acc[v]);
        }
    }
#pragma unroll
    for (int v = 0; v < 8; ++v) {
        float sum = redux16(acc[v]);
        if (m16 == 0)
            rowsum_part[(size_t)chunk * BN + bN + row0 + v + 8 * hf] = sum;
    }
}

// ---------------------------------------------------------------- R1: rsinv = satl/(rowsum+EPS)
__global__ __launch_bounds__(256) void k_r1(const float* __restrict__ rowsum_part,
                                            const float* __restrict__ satl,
                                            float* __restrict__ rsinv) {
    int idx = blockIdx.x * 256 + threadIdx.x;
    if (idx >= BN) return;
    float sum = 0.0f;
#pragma unroll
    for (int c = 0; c < JC; ++c) sum += rowsum_part[(size_t)c * BN + idx];
    rsinv[idx] = satl[idx] / (sum + EPSV);
}

// ---------------------------------------------------------------- pass 2: colsum partials
// colsum[j] = sum_i exp(level*d2) * satr[j] * rsinv[i]
__global__ __launch_bounds__(256) void k_pass2(const float* __restrict__ x1,
                                               const float* __restrict__ x2,
                                               const float* __restrict__ n1,
                                               const float* __restrict__ n2,
                                               const float* __restrict__ satr,
                                               const float* __restrict__ rsinv,
                                               float* __restrict__ colsum_part,
                                               float lv2) {
    __shared__ float sax[CH], say[CH], saz[CH], sln1[CH], srs[CH];
    const int tid = threadIdx.x;
    const int g = blockIdx.x >> 4;
    const int sb = blockIdx.x & 15;
    const int b = g >> 2, chunk = g & 3;
    const size_t bN = (size_t)b * Nn;
    const size_t bM = (size_t)b * Mm;
    const int i0 = chunk * (Nn / IC);

    for (int i = tid; i < CH; i += 256) {         // stage row-chunk operands
        const float* p = x1 + (bN + i0 + i) * 3;
        sax[i] = -2.0f * p[0]; say[i] = -2.0f * p[1]; saz[i] = -2.0f * p[2];
        sln1[i] = lv2 * n1[bN + i0 + i];
        srs[i] = rsinv[bN + i0 + i];
    }
    __syncthreads();

    const int lane = tid & 31;
    const int cs = sb * WPB + (tid >> 5);         // column strip 0..127
    const int col0 = cs * 16;
    const int m16 = lane & 15, hf = lane >> 4;

    const int col = col0 + m16;
    const float* p2 = x2 + (bM + col) * 3;
    v2f bb;
    bb.x = hf ? p2[2]        : p2[0];
    bb.y = hf ? n2[bM + col] : p2[1];
    const float sr = satr[bM + col];

    float colacc = 0.0f;
    for (int t = 0; t < CH / 16; ++t) {
        int rr = t * 16 + m16;
        v2f a;
        a.x = hf ? saz[rr] : sax[rr];
        a.y = hf ? 1.0f    : say[rr];
        v8f c = wmma_d2(a, bb);
#pragma unroll
        for (int v = 0; v < 8; ++v) {
            int rl = t * 16 + v + 8 * hf;
            float e = fast_exp2(fmaf(lv2, c[v], sln1[rl]));
            colacc = fmaf(e, srs[rl], colacc);
        }
    }
    colacc *= sr;                                  // per-column factor hoisted
    colacc += __shfl_xor(colacc, 16);
    if (lane < 16)
        colsum_part[(size_t)chunk * BM + bM + col0 + lane] = colacc;
}

// ---------------------------------------------------------------- R2: r, satr update
__global__ __launch_bounds__(256) void k_r2(const float* __restrict__ colsum_part,
                                            const float* __restrict__ satr_old,
                                            float* __restrict__ rcol,
                                            float* __restrict__ satr_new) {
    int idx = blockIdx.x * 256 + threadIdx.x;
    if (idx >= BM) return;
    float cs = 0.0f;
#pragma unroll
    for (int c = 0; c < IC; ++c) cs += colsum_part[(size_t)c * BM + idx];
    float so = satr_old[idx];
    float r  = fminf(so / (cs + EPSV), 1.0f);
    rcol[idx] = r;
    satr_new[idx] = fmaxf(so - cs * r, 0.0f);
}

// ---------------------------------------------------------------- pass 3: cost + rowsum2 partials
// w = exp(level*d2) * (satr[j]*r[j]) * rsinv[i];  cost += w*sqrt(d2)
__global__ __launch_bounds__(256) void k_pass3(const float* __restrict__ x1,
                                               const float* __restrict__ x2,
                                               const float* __restrict__ n1,
                                               const float* __restrict__ n2,
                                               const float* __restrict__ satr,
                                               const float* __restrict__ rcol,
                                               const float* __restrict__ rsinv,
                                               float* __restrict__ rowsum2_part,
                                               float* __restrict__ cost_part,
                                               float lv2) {
    __shared__ float sx[CH], sy[CH], sz[CH], sn[CH], sw[CH];
    const int tid = threadIdx.x;
    const int g = blockIdx.x >> 4;
    const int sb = blockIdx.x & 15;
    const int b = g >> 2, chunk = g & 3;
    const size_t bN = (size_t)b * Nn;
    const size_t bM = (size_t)b * Mm;
    const int j0 = chunk * CH;

    for (int i = tid; i < CH; i += 256) {
        const float* p = x2 + (bM + j0 + i) * 3;
        sx[i] = p[0]; sy[i] = p[1]; sz[i] = p[2];
        sn[i] = n2[bM + j0 + i];
        sw[i] = satr[bM + j0 + i] * rcol[bM + j0 + i];  // per-column product
    }
    __syncthreads();

    const int lane = tid & 31;
    const int s = sb * WPB + (tid >> 5);
    const int row0 = s * 16;
    const int m16 = lane & 15, hf = lane >> 4;

    const float* p1 = x1 + (bN + row0 + m16) * 3;
    v2f a;
    a.x = hf ? -2.0f * p1[2] : -2.0f * p1[0];
    a.y = hf ? 1.0f          : -2.0f * p1[1];

    float n1v[8], rsl[8];
#pragma unroll
    for (int v = 0; v < 8; ++v) {
        size_t rv = bN + row0 + v + 8 * hf;
        n1v[v] = n1[rv];
        rsl[v] = rsinv[rv];
    }

    float rowacc[8];
#pragma unroll
    for (int v = 0; v < 8; ++v) rowacc[v] = 0.0f;
    float costacc = 0.0f;

    for (int t = 0; t < CH / 16; ++t) {
        int cc = t * 16 + m16;
        v2f bb;
        bb.x = hf ? sz[cc] : sx[cc];
        bb.y = hf ? sn[cc] : sy[cc];
        float swc = sw[cc];
        v8f c = wmma_d2(a, bb);
#pragma unroll
        for (int v = 0; v < 8; ++v) {
            float d2 = c[v] + n1v[v];
            float e  = fast_exp2(lv2 * d2);
            float wt = e * swc * rsl[v];
            rowacc[v] += wt;
            costacc = fmaf(wt, fast_sqrt(fmaxf(d2, 0.0f)), costacc);
        }
    }
#pragma unroll
    for (int v = 0; v < 8; ++v) {
        float sum = redux16(rowacc[v]);
        if (m16 == 0)
            rowsum2_part[(size_t)chunk * BN + bN + row0 + v + 8 * hf] = sum;
    }
    costacc = redux16(costacc);
    costacc += __shfl_xor(costacc, 16);
    if (lane == 0)
        cost_part[((size_t)b * TN + s) * JC + chunk] += costacc;  // owned slot
}

// ---------------------------------------------------------------- R3: satl update
__global__ __launch_bounds__(256) void k_r3(const float* __restrict__ rowsum2_part,
                                            const float* __restrict__ satl_old,
                                            float* __restrict__ satl_new) {
    int idx = blockIdx.x * 256 + threadIdx.x;
    if (idx >= BN) return;
    float sum = 0.0f;
#pragma unroll
    for (int c = 0; c < JC; ++c) sum += rowsum2_part[(size_t)c * BN + idx];
    satl_new[idx] = fmaxf(satl_old[idx] - sum, 0.0f);
}

// ---------------------------------------------------------------- final: per-batch cost
__global__ __launch_bounds__(256) void k_final(const float* __restrict__ cost_part,
                                               float* __restrict__ out) {
    __shared__ float sm[256];
    int b = blockIdx.x, t = threadIdx.x;
    const int per = TN * JC;                       // 512 partials per batch
    float sum = 0.0f;
    for (int i = t; i < per; i += 256) sum += cost_part[(size_t)b * per + i];
    sm[t] = sum;
    __syncthreads();
    for (int o = 128; o > 0; o >>= 1) {
        if (t < o) sm[t] += sm[t + o];
        __syncthreads();
    }
    if (t == 0) out[b] = sm[0];
}

// ---------------------------------------------------------------- host
extern "C" void kernel_launch(void* const* d_in, const int* in_sizes, int n_in,
                              void* d_out, int out_size, void* d_ws, size_t ws_size,
                              hipStream_t stream) {
    (void)in_sizes; (void)n_in; (void)out_size; (void)ws_size;
    const float* x1 = (const float*)d_in[0];
    const float* x2 = (const float*)d_in[1];
    float* out = (float*)d_out;

    float* w = (float*)d_ws;
    float* n1     = w;              w += BN;
    float* n2     = w;              w += BM;
    float* satlA  = w;              w += BN;
    float* satlB  = w;              w += BN;
    float* satrA  = w;              w += BM;
    float* satrB  = w;              w += BM;
    float* rsinv  = w;              w += BN;      // satl/(rowsum+EPS)
    float* rcol   = w;              w += BM;
    float* rsp    = w;              w += (size_t)JC * BN;   // rowsum partials
    float* csp    = w;              w += (size_t)IC * BM;   // colsum partials
    float* rs2p   = w;              w += (size_t)JC * BN;   // rowsum2 partials
    float* costp  = w;              w += (size_t)Bsz * TN * JC;

    const dim3 blk(256);
    const dim3 gridSmall((BN + 255) / 256);                  // 64 blocks
    const dim3 gridBig(Bsz * JC * (TN / WPB));               // 512 blocks

    k_prep<<<gridSmall, blk, 0, stream>>>(x1, x2, n1, n2, satlA, satrA, costp);

    const float levels[11] = {-65536.f, -16384.f, -4096.f, -1024.f, -256.f,
                              -64.f, -16.f, -4.f, -1.f, -0.25f, 0.f};
    const float LOG2E = 1.4426950408889634f;

    float *satl_cur = satlA, *satl_nxt = satlB;
    float *satr_cur = satrA, *satr_nxt = satrB;

    for (int L = 0; L < 11; ++L) {
        float lv2 = levels[L] * LOG2E;
        k_pass1<<<gridBig, blk, 0, stream>>>(x1, x2, n1, n2, satr_cur, rsp, lv2);
        k_r1<<<gridSmall, blk, 0, stream>>>(rsp, satl_cur, rsinv);
        k_pass2<<<gridBig, blk, 0, stream>>>(x1, x2, n1, n2, satr_cur, rsinv,
                                             csp, lv2);
        k_r2<<<gridSmall, blk, 0, stream>>>(csp, satr_cur, rcol, satr_nxt);
        k_pass3<<<gridBig, blk, 0, stream>>>(x1, x2, n1, n2, satr_cur, rcol,
                                             rsinv, rs2p, costp, lv2);
        k_r3<<<gridSmall, blk, 0, stream>>>(rs2p, satl_cur, satl_nxt);
        float* t;
        t = satl_cur; satl_cur = satl_nxt; satl_nxt = t;
        t = satr_cur; satr_cur = satr_nxt; satr_nxt = t;
    }

    k_final<<<dim3(Bsz), blk, 0, stream>>>(costp, out);
}